// Extrema2D_27504970563628
// MI455X (gfx1250) — compile-verified
//
#include <hip/hip_runtime.h>
#include <cmath>

// ---------------------------------------------------------------------------
// Extrema2D + greedy 7x7 NMS (fixed-point relaxation), B=32, H=W=512, f32.
// kept bit-packed 32 pixels/word (wave32 ballot). 20 deterministic sweeps.
// HBM floor ~2.9us (32MiB in + 32MiB out @ 23.3TB/s); the 20 relaxation
// sweeps run out of the 192MB L2 (x: 32MiB + masks: 2MiB resident).
// ---------------------------------------------------------------------------

#define BATCH 32
#define HT    512
#define WD    512
#define WPR   (WD / 32)          // kept words per image row = 16
#define HALO  3
#define TS    32
#define TW    (TS + 2 * HALO)    // 38
#define TP    (TW + 2)           // LDS row pitch 40 (bank-conflict pad)
#define NMS_ITERS 20

#if defined(__has_builtin)
# if __has_builtin(__builtin_amdgcn_global_load_async_to_lds_b32) && \
     __has_builtin(__builtin_amdgcn_s_wait_asynccnt)
#  define HAVE_ASYNC_LDS 1
# endif
#endif
#ifndef HAVE_ASYNC_LDS
# define HAVE_ASYNC_LDS 0
#endif

#if HAVE_ASYNC_LDS
typedef __attribute__((address_space(1))) int GInt;   // global (AS1)
typedef __attribute__((address_space(3))) int LInt;   // LDS (AS3)
#endif

__device__ __forceinline__ unsigned wave_ballot32(bool p) {
#if defined(__has_builtin) && __has_builtin(__builtin_amdgcn_ballot_w32)
    return __builtin_amdgcn_ballot_w32(p);
#else
    return (unsigned)__ballot((int)p);
#endif
}

// Extrema mask exactly mirroring the JAX padding/sign logic:
//  dxR = (w < W-1) && (x[w+1]-x[w] >  0)   (right zero-pad: >0  -> false)
//  dxL = (w == 0)  || (x[w]-x[w-1] <= 0)   (left  zero-pad: <=0 -> true)
//  neg = (x <= 0)
//  e = (valley_x & valley_y) | (peak_x & peak_y)
__device__ __forceinline__ bool extrema_e(float xc, float xl, float xr,
                                          float xu, float xd, int gx, int gy) {
    const bool dxR = (gx < WD - 1) && ((xr - xc) >  0.0f);
    const bool dxL = (gx == 0)     || ((xc - xl) <= 0.0f);
    const bool dyR = (gy < HT - 1) && ((xd - xc) >  0.0f);
    const bool dyL = (gy == 0)     || ((xc - xu) <= 0.0f);
    const bool neg = (xc <= 0.0f);
    const bool valley =  dxR &&  dxL &&  dyR &&  dyL &&  neg;
    const bool peak   = !dxR && !dxL && !dyR && !dyL && !neg;
    return valley || peak;
}

// ---------------------------------------------------------------------------
// NMS relaxation sweep over 32x32 tiles (halo 3) in LDS. (Defined first so
// the disasm snippet shows the async global->LDS + s_wait_asynccnt path.)
// kept_new = e && (|x| >= max over 7x7 of (kept ? |x| : -inf)).
// grid (16, 16, 32), block 256.
// ---------------------------------------------------------------------------
__global__ __launch_bounds__(256) void extrema_nms_iter(
        const float* __restrict__ x,
        const unsigned* __restrict__ ksrc,
        unsigned* __restrict__ kdst) {
    __shared__ float    xs [TW][TP];   // x tile with halo (clamped coords)
    __shared__ float    msk[TW][TP];   // kept ? |x| : -inf
    __shared__ float    hmx[TW][TP];   // horizontal 7-max
    __shared__ unsigned kw [TW][4];    // 3 kept words per tile row

    const int tid  = threadIdx.x;
    const int lane = tid & 31;
    const int wv   = tid >> 5;
    const int b    = blockIdx.z;
    const int x0   = blockIdx.x * TS;
    const int y0   = blockIdx.y * TS;

    const float*    xb = x    + (long)b * HT * WD;
    const unsigned* kb = ksrc + (long)b * HT * WPR;

    // ---- fill x tile (async global -> LDS) --------------------------------
    for (int i = tid; i < TW * TW; i += 256) {
        const int r  = i / TW;
        const int c  = i - r * TW;
        int gy = y0 - HALO + r; gy = gy < 0 ? 0 : (gy > HT - 1 ? HT - 1 : gy);
        int gx = x0 - HALO + c; gx = gx < 0 ? 0 : (gx > WD - 1 ? WD - 1 : gx);
#if HAVE_ASYNC_LDS
        __builtin_amdgcn_global_load_async_to_lds_b32(
            (GInt*)(const void*)(xb + (long)gy * WD + gx),
            (LInt*)(void*)&xs[r][c], 0, 0);
#else
        xs[r][c] = xb[(long)gy * WD + gx];
#endif
    }
    // ---- kept words: 3 per row cover bit-columns [x0-32, x0+63] -----------
    for (int i = tid; i < TW * 3; i += 256) {
        const int r  = i / 3;
        const int wi = i - r * 3;
        const int gy = y0 - HALO + r;
        const int wx = (x0 >> 5) - 1 + wi;
        unsigned v = 0;
        if (gy >= 0 && gy < HT && wx >= 0 && wx < WPR)
            v = kb[(long)gy * WPR + wx];
        kw[r][wi] = v;
    }
#if HAVE_ASYNC_LDS
    __builtin_amdgcn_s_wait_asynccnt(0);
#endif
    __syncthreads();

    // ---- masked scores ----------------------------------------------------
    for (int i = tid; i < TW * TW; i += 256) {
        const int r  = i / TW;
        const int c  = i - r * TW;
        const int bp = c + 29;   // bit-column relative to word (x0/32 - 1)
        const bool k = (kw[r][bp >> 5] >> (bp & 31)) & 1u;
        msk[r][c] = k ? fabsf(xs[r][c]) : -INFINITY;
    }
    __syncthreads();

    // ---- horizontal 7-max -------------------------------------------------
    for (int i = tid; i < TW * TS; i += 256) {
        const int r = i >> 5;
        const int c = (i & 31) + HALO;
        float m = msk[r][c - 3];
#pragma unroll
        for (int d = -2; d <= 3; ++d) m = fmaxf(m, msk[r][c + d]);
        hmx[r][c] = m;
    }
    __syncthreads();

    // ---- vertical 7-max + decision; one ballot word per row ---------------
#pragma unroll
    for (int k = 0; k < 4; ++k) {
        const int ty = wv + 8 * k;
        const int tx = lane;
        const int r  = ty + HALO;
        const int c  = tx + HALO;
        float m = hmx[r - 3][c];
#pragma unroll
        for (int d = -2; d <= 3; ++d) m = fmaxf(m, hmx[r + d][c]);

        const float xc = xs[r][c];
        const int gx = x0 + tx, gy = y0 + ty;
        const bool e = extrema_e(xc, xs[r][c - 1], xs[r][c + 1],
                                 xs[r - 1][c], xs[r + 1][c], gx, gy);
        const bool kept = e && (fabsf(xc) >= m);
        const unsigned w = wave_ballot32(kept);
        if (lane == 0)
            kdst[((long)b * HT + gy) * WPR + (x0 >> 5)] = w;
    }
}

// ---------------------------------------------------------------------------
// kept0 = extrema mask, bit-packed.
// grid (16, 64, 32), block 256 = 8 waves; wave handles one row of 32 pixels.
// ---------------------------------------------------------------------------
__global__ __launch_bounds__(256) void extrema_init(const float* __restrict__ x,
                                                    unsigned* __restrict__ kept) {
    const int lane = threadIdx.x & 31;
    const int wv   = threadIdx.x >> 5;
    const int gx   = blockIdx.x * 32 + lane;
    const int gy   = blockIdx.y * 8 + wv;
    const int b    = blockIdx.z;

    const long base = ((long)b * HT + gy) * WD + gx;
    const float xc = x[base];
    const float xl = (gx > 0)      ? x[base - 1]  : 0.0f;
    const float xr = (gx < WD - 1) ? x[base + 1]  : 0.0f;
    const float xu = (gy > 0)      ? x[base - WD] : 0.0f;
    const float xd = (gy < HT - 1) ? x[base + WD] : 0.0f;

    const bool e = extrema_e(xc, xl, xr, xu, xd, gx, gy);
    const unsigned w = wave_ballot32(e);
    if (lane == 0)
        kept[((long)b * HT + gy) * WPR + blockIdx.x] = w;
}

// ---------------------------------------------------------------------------
// out = kept ? x : 0, vectorized float4 (4 pixels share one kept word).
// ---------------------------------------------------------------------------
__global__ __launch_bounds__(256) void extrema_finalize(
        const float4* __restrict__ x4,
        const unsigned* __restrict__ kept,
        float4* __restrict__ out4) {
    const long q = (long)blockIdx.x * 256 + threadIdx.x;   // quad index
    const long i = q * 4;
    const unsigned w = kept[i >> 5] >> ((int)i & 31);
    float4 v = x4[q];
    v.x = (w & 1u) ? v.x : 0.0f;
    v.y = (w & 2u) ? v.y : 0.0f;
    v.z = (w & 4u) ? v.z : 0.0f;
    v.w = (w & 8u) ? v.w : 0.0f;
    out4[q] = v;
}

// ---------------------------------------------------------------------------
extern "C" void kernel_launch(void* const* d_in, const int* in_sizes, int n_in,
                              void* d_out, int out_size, void* d_ws, size_t ws_size,
                              hipStream_t stream) {
    (void)in_sizes; (void)n_in; (void)out_size; (void)ws_size;
    const float* x = (const float*)d_in[0];
    constexpr long N     = (long)BATCH * HT * WD;       // 8,388,608
    constexpr long WORDS = N / 32;                      // 262,144 (1 MiB/buf)

    unsigned* bufA = (unsigned*)d_ws;
    unsigned* bufB = bufA + WORDS;

    extrema_init<<<dim3(16, 64, 32), 256, 0, stream>>>(x, bufA);

    unsigned* src = bufA;
    unsigned* dst = bufB;
    for (int it = 0; it < NMS_ITERS; ++it) {
        extrema_nms_iter<<<dim3(16, 16, 32), 256, 0, stream>>>(x, src, dst);
        unsigned* t = src; src = dst; dst = t;
    }
    // NMS_ITERS is even -> final kept mask is back in bufA (== src).
    extrema_finalize<<<dim3((unsigned)(N / 4 / 256)), 256, 0, stream>>>(
        (const float4*)x, src, (float4*)d_out);
}